// VoteAttention_8100308320900
// MI455X (gfx1250) — compile-verified
//
#include <hip/hip_runtime.h>
#include <math.h>

typedef __attribute__((ext_vector_type(2))) float v2f;
typedef __attribute__((ext_vector_type(8))) float v8f;

#define NSEEDS 2500
#define NJ     24
#define DF     131
#define HSTR   145            // odd padded stride for H/Q in LDS -> no bank conflicts
#define NWAVES 8
#define NTHREADS (NWAVES * 32)

__device__ __forceinline__ int imin(int a, int b) { return a < b ? a : b; }

// D = A(16x4 f32) * B(4x16 f32) + C(16x16 f32), full f32 precision.
__device__ __forceinline__ v8f wmma4(v2f a, v2f b, v8f c) {
    return __builtin_amdgcn_wmma_f32_16x16x4_f32(
        /*neg_a=*/false, a, /*neg_b=*/false, b,
        /*c_mod=*/(short)0, c, /*reuse_a=*/false, /*reuse_b=*/false);
}

__global__ __launch_bounds__(NTHREADS)
void vote_attention_fused(const float* __restrict__ seed,   // [B,2500,131]
                          const float* __restrict__ W1,     // [24,2500]
                          const float* __restrict__ b1,     // [24]
                          const float* __restrict__ W2,     // [24,24]
                          const float* __restrict__ b2,     // [24]
                          float* __restrict__ out,          // [B,24,131]
                          float* __restrict__ att)          // [B,24,2500]
{
    extern __shared__ float smem[];
    float* W1s = smem;                      // [24][2500], later reused as scores
    float* SC  = smem;                      // alias: scores/attention [24][2500]
    float* Hs  = smem + NJ * NSEEDS;        // [32][HSTR]
    float* Qs  = Hs + 32 * HSTR;            // [32][HSTR]

    const int b    = blockIdx.x;
    const int tid  = threadIdx.x;
    const int wave = tid >> 5;
    const int lane = tid & 31;
    const int half = lane >> 4;             // ISA: lanes 0-15 vs 16-31
    const int l16  = lane & 15;

    const float* S = seed + (size_t)b * NSEEDS * DF;

    // A-tile row indices (clamped addresses; clamped rows' results are
    // discarded by epilogue guards -- WMMA rows are independent).
    const int m0  = l16;                    // rows 0..15, always valid
    const int m1  = 16 + l16;               // rows 16..31, valid 16..23
    const int m1c = imin(m1, NJ - 1);

    // ---- stage W1 into LDS once (reused by all 9 N-tiles) ----
    for (int i = tid; i < NJ * NSEEDS; i += NTHREADS) W1s[i] = W1[i];
    __syncthreads();

    // ============ Stage 1: H = relu(W1 @ S + b1)  [M=24,N=131,K=2500] ========
    for (int nt = wave; nt < 9; nt += NWAVES) {
        const int n0   = nt * 16;
        const int col  = n0 + l16;
        const int colc = imin(col, DF - 1);      // clamped: cols >=131 give garbage
        v8f c0 = {}; v8f c1 = {};                // in discarded output columns only
        #pragma unroll 5
        for (int k0 = 0; k0 < NSEEDS; k0 += 4) {
            const int ka = k0 + 2 * half;
            v2f bt;
            bt.x = S[(size_t)ka * DF + colc];
            bt.y = S[(size_t)(ka + 1) * DF + colc];
            v2f a0; a0.x = W1s[m0  * NSEEDS + ka]; a0.y = W1s[m0  * NSEEDS + ka + 1];
            v2f a1; a1.x = W1s[m1c * NSEEDS + ka]; a1.y = W1s[m1c * NSEEDS + ka + 1];
            c0 = wmma4(a0, bt, c0);
            c1 = wmma4(a1, bt, c1);
        }
        #pragma unroll
        for (int v = 0; v < 8; ++v) {
            const int m  = v + 8 * half;         // C/D layout: row = vgpr + 8*half
            const int m2 = m + 16;
            Hs[m  * HSTR + col] = fmaxf(c0[v] + b1[imin(m,  NJ - 1)], 0.0f);
            Hs[m2 * HSTR + col] = fmaxf(c1[v] + b1[imin(m2, NJ - 1)], 0.0f);
        }
    }
    __syncthreads();

    // ============ Stage 2: Q = W2 @ H + b2  [M=24,N=131,K=24] ================
    for (int nt = wave; nt < 9; nt += NWAVES) {
        const int n0  = nt * 16;
        const int col = n0 + l16;
        v8f c0 = {}; v8f c1 = {};
        #pragma unroll
        for (int k0 = 0; k0 < NJ; k0 += 4) {
            const int ka = k0 + 2 * half;
            v2f bt; bt.x = Hs[ka * HSTR + col]; bt.y = Hs[(ka + 1) * HSTR + col];
            v2f a0; a0.x = W2[m0  * NJ + ka];   a0.y = W2[m0  * NJ + ka + 1];
            v2f a1; a1.x = W2[m1c * NJ + ka];   a1.y = W2[m1c * NJ + ka + 1];
            c0 = wmma4(a0, bt, c0);
            c1 = wmma4(a1, bt, c1);
        }
        #pragma unroll
        for (int v = 0; v < 8; ++v) {
            const int m  = v + 8 * half;
            const int m2 = m + 16;
            Qs[m  * HSTR + col] = c0[v] + b2[imin(m,  NJ - 1)];
            Qs[m2 * HSTR + col] = c1[v] + b2[imin(m2, NJ - 1)];
        }
    }
    __syncthreads();

    // ============ Stage 3: scores = Q @ S^T  [M=24,N=2500,K=131] =============
    // Qs has 32 rows, so A-row addresses are valid unclamped; garbage rows
    // 24..31 produce garbage score rows that the store guard drops.
    for (int nt = wave; nt < 157; nt += NWAVES) {
        const int n0    = nt * 16;
        const int nrow  = n0 + l16;              // seed row feeding this B column
        const int nrowc = imin(nrow, NSEEDS - 1);
        const float* Srow = S + (size_t)nrowc * DF;
        v8f c0 = {}; v8f c1 = {};
        #pragma unroll 4
        for (int k0 = 0; k0 < 128; k0 += 4) {    // guard-free body (k <= 127)
            const int ka = k0 + 2 * half;
            v2f bt; bt.x = Srow[ka];            bt.y = Srow[ka + 1];
            v2f a0; a0.x = Qs[m0 * HSTR + ka];  a0.y = Qs[m0 * HSTR + ka + 1];
            v2f a1; a1.x = Qs[m1 * HSTR + ka];  a1.y = Qs[m1 * HSTR + ka + 1];
            c0 = wmma4(a0, bt, c0);
            c1 = wmma4(a1, bt, c1);
        }
        {   // peeled final step: k = 128..131; mask out k==131 (branchless)
            const int  ka   = 128 + 2 * half;
            const bool kyok = (ka + 1 < DF);     // half0: k=129 ok, half1: k=131 no
            const int  kyc  = imin(ka + 1, DF - 1);
            v2f bt; bt.x = Srow[ka];
                    bt.y = kyok ? Srow[kyc] : 0.0f;
            v2f a0; a0.x = Qs[m0 * HSTR + ka];
                    a0.y = kyok ? Qs[m0 * HSTR + kyc] : 0.0f;
            v2f a1; a1.x = Qs[m1 * HSTR + ka];
                    a1.y = kyok ? Qs[m1 * HSTR + kyc] : 0.0f;
            c0 = wmma4(a0, bt, c0);
            c1 = wmma4(a1, bt, c1);
        }
        const int nc = n0 + l16;
        if (nc < NSEEDS) {
            #pragma unroll
            for (int v = 0; v < 8; ++v) {
                const int m  = v + 8 * half;
                const int m2 = m + 16;
                if (m  < NJ) SC[m  * NSEEDS + nc] = c0[v];
                if (m2 < NJ) SC[m2 * NSEEDS + nc] = c1[v];
            }
        }
    }
    __syncthreads();

    // ============ Stage 4: row softmax (one wave per row, wave32) ============
    for (int r = wave; r < NJ; r += NWAVES) {
        float* row = SC + r * NSEEDS;
        float mx = -INFINITY;
        for (int n = lane; n < NSEEDS; n += 32) mx = fmaxf(mx, row[n]);
        for (int off = 16; off; off >>= 1) mx = fmaxf(mx, __shfl_xor(mx, off, 32));
        float sum = 0.0f;
        for (int n = lane; n < NSEEDS; n += 32) {
            float e = __expf(row[n] - mx);
            row[n] = e;
            sum += e;
        }
        for (int off = 16; off; off >>= 1) sum += __shfl_xor(sum, off, 32);
        const float inv = 1.0f / sum;
        float* grow = att + ((size_t)b * NJ + r) * NSEEDS;
        for (int n = lane; n < NSEEDS; n += 32) {
            float a = row[n] * inv;
            row[n] = a;                              // keep in LDS for stage 5
            __builtin_nontemporal_store(a, grow + n); // write-once: bypass L2 RT
        }
    }
    __syncthreads();

    // ============ Stage 5: out = att @ S  [M=24,N=131,K=2500] ================
    for (int nt = wave; nt < 9; nt += NWAVES) {
        const int n0   = nt * 16;
        const int col  = n0 + l16;
        const int colc = imin(col, DF - 1);
        v8f c0 = {}; v8f c1 = {};
        #pragma unroll 5
        for (int k0 = 0; k0 < NSEEDS; k0 += 4) {
            const int ka = k0 + 2 * half;
            v2f bt;
            bt.x = S[(size_t)ka * DF + colc];
            bt.y = S[(size_t)(ka + 1) * DF + colc];
            v2f a0; a0.x = SC[m0  * NSEEDS + ka]; a0.y = SC[m0  * NSEEDS + ka + 1];
            v2f a1; a1.x = SC[m1c * NSEEDS + ka]; a1.y = SC[m1c * NSEEDS + ka + 1];
            c0 = wmma4(a0, bt, c0);
            c1 = wmma4(a1, bt, c1);
        }
        if (col < DF) {
            #pragma unroll
            for (int v = 0; v < 8; ++v) {
                const int m  = v + 8 * half;
                const int m2 = m + 16;
                if (m  < NJ)
                    __builtin_nontemporal_store(c0[v], out + ((size_t)b * NJ + m ) * DF + col);
                if (m2 < NJ)
                    __builtin_nontemporal_store(c1[v], out + ((size_t)b * NJ + m2) * DF + col);
            }
        }
    }
}

extern "C" void kernel_launch(void* const* d_in, const int* in_sizes, int n_in,
                              void* d_out, int out_size, void* d_ws, size_t ws_size,
                              hipStream_t stream) {
    const float* seed = (const float*)d_in[0];
    const float* W1   = (const float*)d_in[1];
    const float* b1   = (const float*)d_in[2];
    const float* W2   = (const float*)d_in[3];
    const float* b2   = (const float*)d_in[4];

    const int B = in_sizes[0] / (NSEEDS * DF);     // 256

    float* out = (float*)d_out;                    // [B,24,131]
    float* att = out + (size_t)B * NJ * DF;        // [B,24,2500]

    const size_t smem = (size_t)(NJ * NSEEDS + 2 * 32 * HSTR) * sizeof(float); // ~277 KB

    // CDNA5 WGP has 320 KB LDS; raise the dynamic-LDS cap for this kernel.
    (void)hipFuncSetAttribute((const void*)vote_attention_fused,
                              hipFuncAttributeMaxDynamicSharedMemorySize,
                              (int)smem);

    vote_attention_fused<<<dim3(B), dim3(NTHREADS), smem, stream>>>(
        seed, W1, b1, W2, b2, out, att);
}